// GCN2_7730941133132
// MI455X (gfx1250) — compile-verified
//
#include <hip/hip_runtime.h>
#include <hip/hip_bf16.h>

#define NNODES 50000
#define NFEAT  32
#define HIDDEN 256
#define NCLASS 6

typedef __attribute__((ext_vector_type(2))) float v2f;
typedef __attribute__((ext_vector_type(8))) float v8f;

// ---------------- degree / normalization ----------------

__global__ void k_deg_init(float* __restrict__ deg, int n) {
    int i = blockIdx.x * blockDim.x + threadIdx.x;
    if (i < n) deg[i] = 1.0f;              // self-loop contributes 1
}

__global__ void k_deg_count(const int* __restrict__ dst, float* __restrict__ deg, int ne) {
    int e = blockIdx.x * blockDim.x + threadIdx.x;
    if (e < ne) atomicAdd(&deg[dst[e]], 1.0f);
}

__global__ void k_deg_rsqrt(float* __restrict__ deg, int n) {
    int i = blockIdx.x * blockDim.x + threadIdx.x;
    if (i < n) deg[i] = rsqrtf(deg[i]);    // in-place: deg -> dinv
}

// ---------------- layer-1 aggregation on 32-wide input ----------------

// agg[n][f] = dinv[n]^2 * x[n][f]   (self-loop term)
__global__ void k_agg_init(const float* __restrict__ x, const float* __restrict__ dinv,
                           float* __restrict__ agg, int total) {
    int i = blockIdx.x * blockDim.x + threadIdx.x;
    if (i < total) {
        int n = i >> 5;                    // /NFEAT
        float di = dinv[n];
        agg[i] = di * di * x[i];
    }
}

// one wave per edge, lane f = feature
__global__ void k_edge_agg32(const float* __restrict__ x, const int* __restrict__ src,
                             const int* __restrict__ dst, const float* __restrict__ dinv,
                             float* __restrict__ agg, int ne) {
    int e = blockIdx.x * (blockDim.x >> 5) + (threadIdx.x >> 5);
    int f = threadIdx.x & 31;
    if (e >= ne) return;
    int s = src[e], d = dst[e];
    float norm = dinv[s] * dinv[d];
    atomicAdd(&agg[d * NFEAT + f], norm * x[s * NFEAT + f]);
}

// ---------------- GEMM1: h = relu(agg @ W1 + b1), fp32 WMMA ----------------
// A-layout (16x4 f32): lane<16 -> M=lane, K pair {k0,k0+1}; lane>=16 -> K pair {k0+2,k0+3}
// B-layout mirrored with N=lane&15.  C/D: vgpr r -> M = r + 8*(lane>>4), N = n0 + (lane&15)
// One wave per (mtile, group of 4 ntiles): A tile stays in registers, reused 4x.

__global__ void k_gemm1_wmma(const float* __restrict__ agg, const float* __restrict__ W1,
                             const float* __restrict__ b1, float* __restrict__ h) {
    const int mtiles  = NNODES / 16;       // 3125
    const int ngroups = (HIDDEN / 16) / 4; // 4
    int wid  = blockIdx.x * (blockDim.x >> 5) + (threadIdx.x >> 5);
    int lane = threadIdx.x & 31;
    if (wid >= mtiles * ngroups) return;   // wave-uniform
    int mt  = wid >> 2;                    // / ngroups
    int ng  = wid & 3;
    int m0  = mt * 16;
    int l15 = lane & 15;
    int kh  = (lane >> 4) & 1;

    // load the full 16x32 A tile once (8 K-steps of v2f per lane)
    const float* arow = agg + (m0 + l15) * NFEAT + 2 * kh;
    v2f a[8];
#pragma unroll
    for (int t = 0; t < 8; ++t) {
        a[t].x = arow[4 * t];
        a[t].y = arow[4 * t + 1];
    }

#pragma unroll
    for (int j = 0; j < 4; ++j) {
        int n0 = (ng * 4 + j) * 16;
        const float* bcol = W1 + (2 * kh) * HIDDEN + n0 + l15;
        v8f c = {};
#pragma unroll
        for (int t = 0; t < 8; ++t) {
            v2f b;
            b.x = bcol[(4 * t) * HIDDEN];
            b.y = bcol[(4 * t + 1) * HIDDEN];
            c = __builtin_amdgcn_wmma_f32_16x16x4_f32(false, a[t], false, b,
                                                      (short)0, c, false, false);
        }
        float bias = b1[n0 + l15];
        float* hout = h + (m0 + 8 * kh) * HIDDEN + n0 + l15;
#pragma unroll
        for (int r = 0; r < 8; ++r) {
            float v = c[r] + bias;
            hout[r * HIDDEN] = v > 0.0f ? v : 0.0f;
        }
    }
}

// ---------------- GEMM2: p = h @ W2  (N padded 6 -> 16), fp32 WMMA ----------------
// Fused epilogue: also writes out[m][c] = dinv[m]^2 * p[m][c] + b2[c]
// (the self-loop + bias term of layer-2 aggregation).

__global__ void k_gemm2_wmma(const float* __restrict__ h, const float* __restrict__ W2,
                             const float* __restrict__ dinv, const float* __restrict__ b2,
                             float* __restrict__ p, float* __restrict__ out) {
    const int mtiles = NNODES / 16;        // 3125
    int wid  = blockIdx.x * (blockDim.x >> 5) + (threadIdx.x >> 5);
    int lane = threadIdx.x & 31;
    if (wid >= mtiles) return;             // wave-uniform
    int m0  = wid * 16;
    int l15 = lane & 15;
    int kh  = (lane >> 4) & 1;

    // zero-pad B columns >= NCLASS without divergent loads
    float bvalid = (l15 < NCLASS) ? 1.0f : 0.0f;
    int   bcolid = (l15 < NCLASS) ? l15 : 0;

    const float* arow = h + (m0 + l15) * HIDDEN + 2 * kh;
    const float* bcol = W2 + (2 * kh) * NCLASS + bcolid;

    v8f c = {};
#pragma unroll 8
    for (int k0 = 0; k0 < HIDDEN; k0 += 4) {
        v2f a, b;
        a.x = arow[k0];
        a.y = arow[k0 + 1];
        b.x = bvalid * bcol[k0 * NCLASS];
        b.y = bvalid * bcol[(k0 + 1) * NCLASS];
        c = __builtin_amdgcn_wmma_f32_16x16x4_f32(false, a, false, b,
                                                  (short)0, c, false, false);
    }

    if (l15 < NCLASS) {
        float b2v = b2[l15];
        float* pout = p   + (m0 + 8 * kh) * NCLASS + l15;
        float* oout = out + (m0 + 8 * kh) * NCLASS + l15;
#pragma unroll
        for (int r = 0; r < 8; ++r) {
            float pv = c[r];
            float di = dinv[m0 + 8 * kh + r];
            pout[r * NCLASS] = pv;                       // raw message value
            oout[r * NCLASS] = di * di * pv + b2v;       // self-loop + bias
        }
    }
}

// ---------------- layer-2 edge aggregation on 6-wide result ----------------

__global__ void k_edge_agg6(const float* __restrict__ p, const int* __restrict__ src,
                            const int* __restrict__ dst, const float* __restrict__ dinv,
                            float* __restrict__ out, int ne) {
    int e = blockIdx.x * blockDim.x + threadIdx.x;
    if (e >= ne) return;
    int s = src[e], d = dst[e];
    float norm = dinv[s] * dinv[d];
#pragma unroll
    for (int cc = 0; cc < NCLASS; ++cc)
        atomicAdd(&out[d * NCLASS + cc], norm * p[s * NCLASS + cc]);
}

// ---------------- launch ----------------

extern "C" void kernel_launch(void* const* d_in, const int* in_sizes, int n_in,
                              void* d_out, int out_size, void* d_ws, size_t ws_size,
                              hipStream_t stream) {
    const float* x    = (const float*)d_in[0];
    const int*   ei   = (const int*)d_in[1];
    const float* W1   = (const float*)d_in[2];
    const float* b1   = (const float*)d_in[3];
    const float* W2   = (const float*)d_in[4];
    const float* b2   = (const float*)d_in[5];
    float*       out  = (float*)d_out;

    const int n  = in_sizes[0] / NFEAT;    // 50000
    const int ne = in_sizes[1] / 2;        // 800000
    const int* src = ei;
    const int* dst = ei + ne;

    // workspace layout (floats), 256-aligned
    float* ws   = (float*)d_ws;
    float* dinv = ws;                              // 50000   -> 50176
    float* agg  = ws + 50176;                      // 1,600,000
    float* h    = ws + 50176 + 1600000;            // 12,800,000
    float* p    = ws + 50176 + 1600000 + 12800000; // 300,000

    // 1) degrees -> dinv
    k_deg_init <<<(n + 255) / 256, 256, 0, stream>>>(dinv, n);
    k_deg_count<<<(ne + 255) / 256, 256, 0, stream>>>(dst, dinv, ne);
    k_deg_rsqrt<<<(n + 255) / 256, 256, 0, stream>>>(dinv, n);

    // 2) layer-1 aggregation on x (32-wide)
    k_agg_init <<<(n * NFEAT + 255) / 256, 256, 0, stream>>>(x, dinv, agg, n * NFEAT);
    k_edge_agg32<<<(ne + 7) / 8, 256, 0, stream>>>(x, src, dst, dinv, agg, ne);

    // 3) h = relu(agg @ W1 + b1)   [fp32 WMMA, A kept in regs across 4 N-tiles]
    {
        int waves = (NNODES / 16) * ((HIDDEN / 16) / 4);   // 12500 waves
        k_gemm1_wmma<<<(waves + 7) / 8, 256, 0, stream>>>(agg, W1, b1, h);
    }

    // 4) p = h @ W2 ; out = dinv^2*p + b2  [fp32 WMMA, fused epilogue]
    {
        int waves = NNODES / 16;                           // 3125 waves
        k_gemm2_wmma<<<(waves + 7) / 8, 256, 0, stream>>>(h, W2, dinv, b2, p, out);
    }

    // 5) layer-2 edge aggregation (6-wide)
    k_edge_agg6<<<(ne + 255) / 256, 256, 0, stream>>>(p, src, dst, dinv, out, ne);
}